// GATLayer_37598143709241
// MI455X (gfx1250) — compile-verified
//
#include <hip/hip_runtime.h>

#define Bq   4
#define Nn   2048
#define IND  256
#define Hh   4
#define Dd   64
#define HD   256
#define Mtot (Bq*Nn)        // 8192 rows total
#define DT   128            // dst-columns per attention block
#define ALPHA 0.2f

typedef float v2f __attribute__((ext_vector_type(2)));
typedef float v8f __attribute__((ext_vector_type(8)));

__device__ __forceinline__ float lrelu(float x) { return x > 0.f ? x : ALPHA * x; }

// ---------------------------------------------------------------------------
// Kernel 1: feat[M,256] = X[M,256] @ W[256,256]   (fp32 WMMA 16x16x4)
// One wave computes a 16(M) x 64(N) strip: 4 accumulators, K in steps of 4.
// ---------------------------------------------------------------------------
__global__ void k_feat_gemm(const float* __restrict__ X,
                            const float* __restrict__ W,
                            float* __restrict__ feat) {
  int gw   = blockIdx.x * 8 + (threadIdx.x >> 5);   // 2048 waves total
  int lane = threadIdx.x & 31;
  int lo = lane & 15, hi = lane >> 4;
  int n_base = (gw & 3) * 64;
  int m_base = (gw >> 2) * 16;

  v8f acc0 = {}, acc1 = {}, acc2 = {}, acc3 = {};
  const float* Arow = X + (size_t)(m_base + lo) * IND;

  for (int k0 = 0; k0 < IND; k0 += 4) {
    // A fragment 16x4: v0 = K=k0 (lanes0-15) / K=k0+2 (lanes16-31); v1 = +1
    v2f a;
    a.x = Arow[k0 + hi * 2];
    a.y = Arow[k0 + hi * 2 + 1];
    const float* Wr0 = W + (size_t)(k0 + hi * 2) * HD + n_base + lo;
    const float* Wr1 = Wr0 + HD;
    v2f b0; b0.x = Wr0[0];  b0.y = Wr1[0];
    v2f b1; b1.x = Wr0[16]; b1.y = Wr1[16];
    v2f b2; b2.x = Wr0[32]; b2.y = Wr1[32];
    v2f b3; b3.x = Wr0[48]; b3.y = Wr1[48];
    acc0 = __builtin_amdgcn_wmma_f32_16x16x4_f32(false, a, false, b0, (short)0, acc0, false, false);
    acc1 = __builtin_amdgcn_wmma_f32_16x16x4_f32(false, a, false, b1, (short)0, acc1, false, false);
    acc2 = __builtin_amdgcn_wmma_f32_16x16x4_f32(false, a, false, b2, (short)0, acc2, false, false);
    acc3 = __builtin_amdgcn_wmma_f32_16x16x4_f32(false, a, false, b3, (short)0, acc3, false, false);
  }
  // C layout: VGPR r -> M = r (lanes0-15) / r+8 (lanes16-31), N = lane&15
  for (int r = 0; r < 8; ++r) {
    int row = m_base + r + hi * 8;
    float* o = feat + (size_t)row * HD + n_base + lo;
    o[0]  = acc0[r];
    o[16] = acc1[r];
    o[32] = acc2[r];
    o[48] = acc3[r];
  }
}

// ---------------------------------------------------------------------------
// Kernel 2: el[m,h] = feat[m,h,:]·attn_l[h,:],  er likewise.
// ---------------------------------------------------------------------------
__global__ void k_scores(const float* __restrict__ feat,
                         const float* __restrict__ al,
                         const float* __restrict__ ar,
                         float* __restrict__ el, float* __restrict__ er) {
  int t = blockIdx.x * 256 + threadIdx.x;   // [0, Mtot*H)
  int m = t >> 2, h = t & 3;
  const float* f = feat + (size_t)m * HD + h * Dd;
  const float* L = al + h * Dd;
  const float* R = ar + h * Dd;
  float sl = 0.f, sr = 0.f;
  #pragma unroll 8
  for (int e = 0; e < Dd; ++e) { float v = f[e]; sl += v * L[e]; sr += v * R[e]; }
  el[t] = sl;
  er[t] = sr;
}

// ---------------------------------------------------------------------------
// Kernel 3: masked column softmax (over s) + attention write.
// Block = (b, 128 dst columns). el cached in LDS (32KB); adj packed to bit
// masks via wave32 ballot (32KB) so adj is read from HBM exactly once.
// ---------------------------------------------------------------------------
__global__ void k_attn(const int* __restrict__ adj,
                       const float* __restrict__ el,
                       const float* __restrict__ er,
                       float* __restrict__ att) {
  __shared__ float    el_s[Nn * Hh];          // 32 KB
  __shared__ unsigned mbits[Nn * (DT / 32)];  // 32 KB
  int tid = threadIdx.x;
  int b   = blockIdx.x / (Nn / DT);
  int d0  = (blockIdx.x % (Nn / DT)) * DT;

  for (int i = tid; i < Nn * Hh; i += 256)
    el_s[i] = el[(size_t)b * Nn * Hh + i];

  // Pack adjacency column mask: 2 source rows per iteration, ballot per wave.
  for (int it = 0; it < Nn / 2; ++it) {
    int s  = it * 2 + (tid >> 7);
    int dl = tid & 127;
    int a  = adj[((size_t)b * Nn + s) * Nn + d0 + dl];
    unsigned mb = __builtin_amdgcn_ballot_w32(a > 0);
    if ((tid & 31) == 0) mbits[s * 4 + ((tid >> 5) & 3)] = mb;
  }
  __syncthreads();

  // Each thread owns two (d,h) columns: c = d_local*4 + h, c0=tid, c1=tid+256
  int c0 = tid, c1 = tid + 256;
  int dl0 = c0 >> 2, h0 = c0 & 3;
  int dl1 = c1 >> 2, h1 = c1 & 3;
  float er0 = er[((size_t)b * Nn + d0 + dl0) * Hh + h0];
  float er1 = er[((size_t)b * Nn + d0 + dl1) * Hh + h1];

  float m0 = -__builtin_inff(), m1 = -__builtin_inff();
  float s0 = 0.f, s1 = 0.f;
  for (int s = 0; s < Nn; ++s) {
    unsigned w0 = mbits[s * 4 + (dl0 >> 5)];
    unsigned w1 = mbits[s * 4 + (dl1 >> 5)];
    if ((w0 >> (dl0 & 31)) & 1u) {
      float x = lrelu(el_s[s * 4 + h0] + er0);
      if (x > m0) { s0 = s0 * __expf(m0 - x) + 1.f; m0 = x; }
      else        { s0 += __expf(x - m0); }
    }
    if ((w1 >> (dl1 & 31)) & 1u) {
      float x = lrelu(el_s[s * 4 + h1] + er1);
      if (x > m1) { s1 = s1 * __expf(m1 - x) + 1.f; m1 = x; }
      else        { s1 += __expf(x - m1); }
    }
  }
  float i0 = s0 > 0.f ? 1.f / s0 : 0.f;
  float i1 = s1 > 0.f ? 1.f / s1 : 0.f;

  // Coalesced writes: per source row s, 512 contiguous floats (d,h) tile.
  for (int s = 0; s < Nn; ++s) {
    size_t base = ((size_t)b * Nn + s) * ((size_t)Nn * Hh) + (size_t)d0 * Hh;
    unsigned w0 = mbits[s * 4 + (dl0 >> 5)];
    unsigned w1 = mbits[s * 4 + (dl1 >> 5)];
    float v0 = 0.f, v1 = 0.f;
    if ((w0 >> (dl0 & 31)) & 1u)
      v0 = __expf(lrelu(el_s[s * 4 + h0] + er0) - m0) * i0;
    if ((w1 >> (dl1 & 31)) & 1u)
      v1 = __expf(lrelu(el_s[s * 4 + h1] + er1) - m1) * i1;
    att[base + c0] = v0;
    att[base + c1] = v1;
  }
}

// ---------------------------------------------------------------------------
// Kernel 4: out[b,s,h,:] = sum_d att[b,s,d,h] * feat[b,d,h,:]
// Per (b,h): 2048x2048x64 fp32 GEMM via WMMA 16x16x4; wave = 16(s) x 64(e).
// ---------------------------------------------------------------------------
__global__ void k_out_gemm(const float* __restrict__ att,
                           const float* __restrict__ feat,
                           float* __restrict__ out) {
  int gw   = blockIdx.x * 8 + (threadIdx.x >> 5);   // 2048 waves
  int lane = threadIdx.x & 31;
  int lo = lane & 15, hi = lane >> 4;
  int s_tile = gw & 127;
  int bh     = gw >> 7;
  int b = bh >> 2, h = bh & 3;
  int s_base = s_tile * 16;

  const float* A  = att + ((size_t)b * Nn + s_base + lo) * ((size_t)Nn * Hh) + h;
  const float* Bm = feat + (size_t)b * Nn * HD + h * Dd + lo;

  v8f acc0 = {}, acc1 = {}, acc2 = {}, acc3 = {};
  for (int k0 = 0; k0 < Nn; k0 += 4) {
    if (k0 + 64 < Nn) __builtin_prefetch(A + (size_t)(k0 + 64) * Hh, 0, 0);
    int k = k0 + hi * 2;
    v2f a;
    a.x = A[(size_t)k * Hh];
    a.y = A[(size_t)(k + 1) * Hh];
    const float* B0 = Bm + (size_t)k * HD;
    const float* B1 = B0 + HD;
    v2f b0; b0.x = B0[0];  b0.y = B1[0];
    v2f b1; b1.x = B0[16]; b1.y = B1[16];
    v2f b2; b2.x = B0[32]; b2.y = B1[32];
    v2f b3; b3.x = B0[48]; b3.y = B1[48];
    acc0 = __builtin_amdgcn_wmma_f32_16x16x4_f32(false, a, false, b0, (short)0, acc0, false, false);
    acc1 = __builtin_amdgcn_wmma_f32_16x16x4_f32(false, a, false, b1, (short)0, acc1, false, false);
    acc2 = __builtin_amdgcn_wmma_f32_16x16x4_f32(false, a, false, b2, (short)0, acc2, false, false);
    acc3 = __builtin_amdgcn_wmma_f32_16x16x4_f32(false, a, false, b3, (short)0, acc3, false, false);
  }
  for (int r = 0; r < 8; ++r) {
    int s = s_base + r + hi * 8;
    float* o = out + ((size_t)b * Nn + s) * HD + h * Dd + lo;
    o[0]  = acc0[r];
    o[16] = acc1[r];
    o[32] = acc2[r];
    o[48] = acc3[r];
  }
}

// ---------------------------------------------------------------------------
extern "C" void kernel_launch(void* const* d_in, const int* in_sizes, int n_in,
                              void* d_out, int out_size, void* d_ws, size_t ws_size,
                              hipStream_t stream) {
  const float* X   = (const float*)d_in[0];   // node_feat [B,N,256]
  const int*   adj = (const int*)  d_in[1];   // adj [B,N,N]
  const float* W   = (const float*)d_in[2];   // W [256,256]
  const float* al  = (const float*)d_in[3];   // attn_l [1,H,D]
  const float* ar  = (const float*)d_in[4];   // attn_r [1,H,D]

  float* out = (float*)d_out;                       // output [B,N,HD]
  float* att = out + (size_t)Bq * Nn * HD;          // attention [B,N,N,H]

  float* feat = (float*)d_ws;                       // [Mtot, HD]  8 MB
  float* el   = feat + (size_t)Mtot * HD;           // [Mtot, H]
  float* er   = el + (size_t)Mtot * Hh;             // [Mtot, H]

  k_feat_gemm<<<256, 256, 0, stream>>>(X, W, feat);
  k_scores  <<<(Mtot * Hh) / 256, 256, 0, stream>>>(feat, al, ar, el, er);
  k_attn    <<<Bq * (Nn / DT), 256, 0, stream>>>(adj, el, er, att);
  k_out_gemm<<<256, 256, 0, stream>>>(att, feat, out);
}